// Sinkhorn_51445118271651
// MI455X (gfx1250) — compile-verified
//
#include <hip/hip_runtime.h>
#include <hip/hip_bf16.h>

// Sinkhorn on s[64,1024,1024] fp32.
// Factorized form: s_t = (s0+eps) * R[i] * C[j]; each iteration is a GEMV
// against s0, executed with V_WMMA_F32_16X16X4_F32 broadcast-reductions.
// R/C carry the validity masks as zeros, so all sums are unmasked streams.

#define EPS   1e-4f
#define NB    64
#define NN    1024

typedef __attribute__((ext_vector_type(2))) float v2f;
typedef __attribute__((ext_vector_type(8))) float v8f;

// ---------------------------------------------------------------- init R ----
__global__ void sink_init_R(float* __restrict__ R, const int* __restrict__ nrows) {
  int idx = blockIdx.x * blockDim.x + threadIdx.x;   // 0 .. 65535
  int b = idx >> 10;
  int i = idx & (NN - 1);
  R[idx] = (i < nrows[b]) ? 1.0f : 0.0f;
}

// ---------------------------------------------------- even iters: update C --
// C[b][j] = (j < ncols[b]) ? 1 / sum_i (S[b][i][j]+eps) * R[b][i] : 0
// One block = one (batch, 16-col strip); 8 waves split the 1024-row K range.
__global__ void sink_colpass(const float* __restrict__ S,
                             const float* __restrict__ R,
                             float* __restrict__ C,
                             const int* __restrict__ ncols) {
  __shared__ float red[8 * 16];
  const int b     = blockIdx.x >> 6;
  const int jbase = (blockIdx.x & 63) << 4;
  const int wave  = threadIdx.x >> 5;
  const int lane  = threadIdx.x & 31;
  const int k0    = (lane < 16) ? 0 : 2;       // K pair held by this lane half
  const int jcol  = jbase + (lane & 15);       // N index held by this lane
  const float* Sb = S + (size_t)b * (NN * NN);
  const float* Rb = R + b * NN;

  v8f acc = {0.f, 0.f, 0.f, 0.f, 0.f, 0.f, 0.f, 0.f};
  const int i0 = wave * 128;
  #pragma unroll 4
  for (int i = i0; i < i0 + 128; i += 4) {
    const float* p = Sb + (i + k0) * NN + jcol;
    __builtin_prefetch(p + 32 * NN, 0, 1);     // global_prefetch_b8, 32 rows ahead
    // B-matrix (4x16): B[k][n] = S[i+k][jbase+n] + eps
    float b0 = p[0]  + EPS;
    float b1 = p[NN] + EPS;
    // A-matrix (16x4): A[m][k] = R[i+k]  (broadcast over M)
    float a0 = Rb[i + k0];
    float a1 = Rb[i + k0 + 1];
    v2f av = {a0, a1};
    v2f bv = {b0, b1};
    // D[m][n] = sum_k R[i+k] * (S[i+k][jbase+n]+eps)   (same for all m)
    acc = __builtin_amdgcn_wmma_f32_16x16x4_f32(
        /*neg_a=*/false, av, /*neg_b=*/false, bv,
        /*c_mod=*/(short)0, acc, /*reuse_a=*/false, /*reuse_b=*/false);
  }
  // every lane's acc[0] holds the partial u[jbase + (lane&15)]
  if (lane < 16) red[wave * 16 + lane] = acc[0];
  __syncthreads();
  if (threadIdx.x < 16) {
    float s = 0.f;
    #pragma unroll
    for (int w = 0; w < 8; ++w) s += red[w * 16 + threadIdx.x];
    int j = jbase + threadIdx.x;
    C[b * NN + j] = (j < ncols[b]) ? (1.0f / s) : 0.0f;
  }
}

// ----------------------------------------------------- odd iters: update R --
// R[b][i] = (i < nrows[b]) ? 1 / sum_j (S[b][i][j]+eps) * C[b][j] : 0
// One block = one (batch, 16-row strip); 8 waves split the 1024-col K range.
__global__ void sink_rowpass(const float* __restrict__ S,
                             const float* __restrict__ C,
                             float* __restrict__ R,
                             const int* __restrict__ nrows) {
  __shared__ float red[8 * 16];
  const int b     = blockIdx.x >> 6;
  const int rbase = (blockIdx.x & 63) << 4;
  const int wave  = threadIdx.x >> 5;
  const int lane  = threadIdx.x & 31;
  const int k0    = (lane < 16) ? 0 : 2;
  const int row   = rbase + (lane & 15);       // M index held by this lane
  const float* Sb = S + (size_t)b * (NN * NN) + row * NN;
  const float* Cb = C + b * NN;

  v8f acc = {0.f, 0.f, 0.f, 0.f, 0.f, 0.f, 0.f, 0.f};
  const int j0 = wave * 128;
  #pragma unroll 4
  for (int j = j0; j < j0 + 128; j += 4) {
    __builtin_prefetch(Sb + j + k0 + 64, 0, 1);
    // A-matrix (16x4): A[m][k] = S[rbase+m][j+k] + eps  (2 contiguous floats/lane)
    float a0 = Sb[j + k0]     + EPS;
    float a1 = Sb[j + k0 + 1] + EPS;
    // B-matrix (4x16): B[k][n] = C[j+k]  (broadcast over N)
    float b0 = Cb[j + k0];
    float b1 = Cb[j + k0 + 1];
    v2f av = {a0, a1};
    v2f bv = {b0, b1};
    // D[m][n] = sum_k (S[rbase+m][j+k]+eps) * C[j+k]   (same for all n)
    acc = __builtin_amdgcn_wmma_f32_16x16x4_f32(
        false, av, false, bv, (short)0, acc, false, false);
  }
  // lane 0 holds rows rbase+0..7 in acc[0..7]; lane 16 holds rows rbase+8..15
  if ((lane & 15) == 0) {
    int half = (lane >> 4) << 3;
    #pragma unroll
    for (int g = 0; g < 8; ++g) red[wave * 16 + half + g] = acc[g];
  }
  __syncthreads();
  if (threadIdx.x < 16) {
    float s = 0.f;
    #pragma unroll
    for (int w = 0; w < 8; ++w) s += red[w * 16 + threadIdx.x];
    int i = rbase + threadIdx.x;
    R[b * NN + i] = (i < nrows[b]) ? (1.0f / s) : 0.0f;
  }
}

// ------------------------------------------------------------- final pass --
// out = (S+eps) * R[i] * C[j]   (zeros outside the valid block fall out of R/C)
__global__ void sink_final(const float* __restrict__ S,
                           const float* __restrict__ R,
                           const float* __restrict__ C,
                           float* __restrict__ out) {
  int idx  = blockIdx.x * blockDim.x + threadIdx.x;  // one float4 per thread
  int j4   = idx & 255;          // 256 float4 per row
  int rowg = idx >> 8;           // b*1024 + i
  int b    = rowg >> 10;
  float4 v = reinterpret_cast<const float4*>(S)[idx];
  float  r = R[rowg];
  float4 c = reinterpret_cast<const float4*>(C)[(b << 8) + j4];
  float4 o;
  o.x = (v.x + EPS) * r * c.x;
  o.y = (v.y + EPS) * r * c.y;
  o.z = (v.z + EPS) * r * c.z;
  o.w = (v.w + EPS) * r * c.w;
  reinterpret_cast<float4*>(out)[idx] = o;
}

// ---------------------------------------------------------------- launch ----
extern "C" void kernel_launch(void* const* d_in, const int* in_sizes, int n_in,
                              void* d_out, int out_size, void* d_ws, size_t ws_size,
                              hipStream_t stream) {
  const float* S     = (const float*)d_in[0];
  // d_in[1] = n1, d_in[2] = n2 (static 1024, unused)
  const int*   nrows = (const int*)d_in[3];
  const int*   ncols = (const int*)d_in[4];
  float*       out   = (float*)d_out;

  float* R = (float*)d_ws;        // [64][1024]
  float* C = R + NB * NN;         // [64][1024]

  sink_init_R<<<(NB * NN) / 256, 256, 0, stream>>>(R, nrows);

  // MAX_ITER = 10: 5x (col-normalize -> C, row-normalize -> R)
  for (int it = 0; it < 5; ++it) {
    sink_colpass<<<NB * 64, 256, 0, stream>>>(S, R, C, ncols);
    sink_rowpass<<<NB * 64, 256, 0, stream>>>(S, C, R, nrows);
  }

  sink_final<<<(NB * NN * NN) / 4 / 256, 256, 0, stream>>>(S, R, C, out);
}